// Soft_Att_67473936220377
// MI455X (gfx1250) — compile-verified
//
#include <hip/hip_runtime.h>
#include <hip/hip_bf16.h>
#include <math.h>

#define BATCH 128
#define TLEN  2048
#define DIM   256
#define NWAVE 8
#define TILE_T 16
#define TILES_PER_WAVE (TLEN / TILE_T / NWAVE)  // 16

typedef __attribute__((ext_vector_type(2))) float v2f;
typedef __attribute__((ext_vector_type(4))) float v4f;
typedef __attribute__((ext_vector_type(8))) float v8f;

__global__ __launch_bounds__(256)
void soft_att_online_kernel(const float* __restrict__ h,
                            const float* __restrict__ h_traj,
                            const unsigned char* __restrict__ mask,
                            const float* __restrict__ w,
                            const float* __restrict__ bias_p,
                            float* __restrict__ out)
{
    __shared__ __align__(16) float s_hw[DIM];   // (h*w) per batch row
    __shared__ float s_m[NWAVE];
    __shared__ float s_l[NWAVE];
    __shared__ float s_z[DIM];

    const int tid  = threadIdx.x;
    const int wave = tid >> 5;
    const int lane = tid & 31;
    const int b    = blockIdx.x;
    const float bias = bias_p[0];

    // stage h[b]*w into LDS; zero the cross-wave z accumulator
    s_hw[tid] = h[(size_t)b * DIM + tid] * w[tid];
    s_z[tid]  = 0.0f;
    __syncthreads();

    const int tIn  = lane & 15;       // column (timestep within tile)
    const int offA = (lane >> 4) * 2; // K sub-offset: 0 for lanes 0-15, 2 for 16-31

    float m_run = -INFINITY;
    float l_run = 0.0f;
    float z[8];
    #pragma unroll
    for (int k = 0; k < 8; ++k) z[k] = 0.0f;

    const size_t rowB = (size_t)b * TLEN;

    for (int it = 0; it < TILES_PER_WAVE; ++it) {
        const int t_base = (it * NWAVE + wave) * TILE_T;

        // ---- scores for 16 timesteps via V_WMMA_F32_16X16X4_F32 ----
        // A (16x4, rows replicated) = hw[k0 .. k0+3]
        //   lanes 0-15 : v0 = hw[k0+0], v1 = hw[k0+1]
        //   lanes 16-31: v0 = hw[k0+2], v1 = hw[k0+3]
        // B (4x16)     = h_traj[b, t_base+n, k0+k]
        //   lanes 0-15 : v0 = x[t=n, k0+0], v1 = x[t=n, k0+1]
        //   lanes 16-31: v0 = x[t=n, k0+2], v1 = x[t=n, k0+3]
        const float* xrow = h_traj + (rowB + t_base + tIn) * DIM + offA;

        v8f c = {};
        #pragma unroll 8
        for (int k0 = 0; k0 < DIM; k0 += 4) {
            v2f a    = *(const v2f*)(s_hw + k0 + offA);   // LDS broadcast
            v2f bmat = *(const v2f*)(xrow + k0);          // global_load_b64
            c = __builtin_amdgcn_wmma_f32_16x16x4_f32(
                    false, a, false, bmat, (short)0, c, false, false);
        }

        // all 16 rows of C identical -> lane holds score for t = t_base + tIn
        float s = c[0] + bias;
        const int t = t_base + tIn;
        if (mask[rowB + t]) s = -INFINITY;

        // tile max across lanes (16 distinct values, duplicated in half-waves)
        float mt = s;
        #pragma unroll
        for (int i = 16; i >= 1; i >>= 1)
            mt = fmaxf(mt, __shfl_xor(mt, i, 32));

        const float m_new = fmaxf(m_run, mt);
        const float scale = (m_run == -INFINITY) ? 0.0f : __expf(m_run - m_new);
        const float e     = (s == -INFINITY) ? 0.0f : __expf(s - m_new);

        // sum of e over the 16 distinct timesteps (zero the duplicate half)
        float ec = (lane < 16) ? e : 0.0f;
        #pragma unroll
        for (int i = 16; i >= 1; i >>= 1)
            ec += __shfl_xor(ec, i, 32);
        l_run = l_run * scale + ec;
        m_run = m_new;

        // ---- online z update: z[d] = z[d]*scale + sum_t e[t]*x[t,d] ----
        #pragma unroll
        for (int k = 0; k < 8; ++k) z[k] *= scale;

        // lane owns d = lane*8 .. lane*8+7 (wave covers contiguous 1 KB rows,
        // re-read hits WGP$/L2 since the score pass just streamed this tile)
        const float* zrow = h_traj + (rowB + t_base) * DIM + lane * 8;
        #pragma unroll 4
        for (int tt = 0; tt < TILE_T; ++tt) {
            const float et = __shfl(e, tt, 32);   // broadcast e[t] from lane tt
            v4f x0 = *(const v4f*)(zrow + (size_t)tt * DIM);
            v4f x1 = *(const v4f*)(zrow + (size_t)tt * DIM + 4);
            z[0] = fmaf(et, x0.x, z[0]);
            z[1] = fmaf(et, x0.y, z[1]);
            z[2] = fmaf(et, x0.z, z[2]);
            z[3] = fmaf(et, x0.w, z[3]);
            z[4] = fmaf(et, x1.x, z[4]);
            z[5] = fmaf(et, x1.y, z[5]);
            z[6] = fmaf(et, x1.z, z[6]);
            z[7] = fmaf(et, x1.w, z[7]);
        }
    }

    // ---- merge the 8 waves' (m, l, z) partial states ----
    if (lane == 0) { s_m[wave] = m_run; s_l[wave] = l_run; }
    __syncthreads();

    float M = -INFINITY;
    #pragma unroll
    for (int wv = 0; wv < NWAVE; ++wv) M = fmaxf(M, s_m[wv]);
    float L = 0.0f;
    #pragma unroll
    for (int wv = 0; wv < NWAVE; ++wv) {
        const float mw = s_m[wv];
        L += (mw == -INFINITY) ? 0.0f : s_l[wv] * __expf(mw - M);
    }

    const float cw = (m_run == -INFINITY) ? 0.0f : __expf(m_run - M);
    #pragma unroll
    for (int k = 0; k < 8; ++k)
        atomicAdd(&s_z[lane * 8 + k], cw * z[k]);   // ds_add_f32
    __syncthreads();

    out[(size_t)b * DIM + tid] = s_z[tid] * (1.0f / L);
}

extern "C" void kernel_launch(void* const* d_in, const int* in_sizes, int n_in,
                              void* d_out, int out_size, void* d_ws, size_t ws_size,
                              hipStream_t stream) {
    const float*         h      = (const float*)d_in[0];          // [B, D]
    const float*         h_traj = (const float*)d_in[1];          // [B, T, D]
    const unsigned char* mask   = (const unsigned char*)d_in[2];  // [B, T] bool
    const float*         w      = (const float*)d_in[3];          // [D]
    const float*         bias   = (const float*)d_in[4];          // scalar
    float*               out    = (float*)d_out;                  // [B, D]

    soft_att_online_kernel<<<BATCH, 256, 0, stream>>>(h, h_traj, mask, w, bias, out);
}